// NeuralDictionaryV5_19430432047761
// MI455X (gfx1250) — compile-verified
//
#include <hip/hip_runtime.h>

#define CAPACITY   65536
#define IN_FEAT    1024
#define OUT_FEAT   1024

#define CHUNKS     128           // row-chunks for the WMMA GEMV
#define CHUNK_ROWS (CAPACITY / CHUNKS)   // 512

typedef float v2f __attribute__((ext_vector_type(2)));
typedef float v8f __attribute__((ext_vector_type(8)));

// ---------------- workspace layout (in floats) ----------------
#define OFF_W      0            // 65536  : scores -> exp weights (in place)
#define OFF_BMAX   65536        // 8192   : per-block maxes from kernel 1
#define OFF_M      73728        // 1      : global max
#define OFF_PSUM   73732        // 256    : per-block exp partial sums
#define OFF_INV    73988        // 1      : 1 / sum
#define OFF_PART   74240        // 131072 : 128 chunks x 1024 GEMV partials

// =====================================================================
// K1: scores[row] = -sum_i |keys[row,i] - q[i]|  ; one wave per row.
//     Also produces a per-block max of the 8 row scores.
// =====================================================================
__global__ void k1_scores(const float* __restrict__ q,
                          const float* __restrict__ keys,
                          float* __restrict__ scores,
                          float* __restrict__ bmax) {
    __shared__ float qs[IN_FEAT];
    __shared__ float wscore[8];
    const int tid  = threadIdx.x;
    // 256 threads x float4 = 1024 floats of query into LDS
    ((float4*)qs)[tid] = ((const float4*)q)[tid];
    __syncthreads();

    const int wave = tid >> 5;
    const int lane = tid & 31;
    const int row  = blockIdx.x * 8 + wave;
    const float* kp = keys + (size_t)row * IN_FEAT;

    float sum = 0.0f;
#pragma unroll
    for (int it = 0; it < 8; ++it) {
        const int colb = it * 128 + lane * 4;   // lane-contiguous float4 -> coalesced b128
        const float4 k  = *(const float4*)(kp + colb);
        const float4 qq = *(const float4*)(qs + colb);
        sum += fabsf(k.x - qq.x) + fabsf(k.y - qq.y)
             + fabsf(k.z - qq.z) + fabsf(k.w - qq.w);
    }
    // wave32 reduction
#pragma unroll
    for (int off = 16; off > 0; off >>= 1)
        sum += __shfl_xor(sum, off, 32);

    if (lane == 0) {
        const float s = -sum;
        scores[row]  = s;
        wscore[wave] = s;
    }
    __syncthreads();
    if (tid == 0) {
        float m = wscore[0];
#pragma unroll
        for (int i = 1; i < 8; ++i) m = fmaxf(m, wscore[i]);
        bmax[blockIdx.x] = m;
    }
}

// =====================================================================
// K2: reduce 8192 per-block maxes -> global max M  (single block)
// =====================================================================
__global__ void k2_maxreduce(const float* __restrict__ bmax,
                             float* __restrict__ Mout) {
    __shared__ float red[1024];
    const int tid = threadIdx.x;
    float m = -INFINITY;
    for (int i = tid; i < 8192; i += 1024) m = fmaxf(m, bmax[i]);
    red[tid] = m;
    __syncthreads();
    for (int s = 512; s > 0; s >>= 1) {
        if (tid < s) red[tid] = fmaxf(red[tid], red[tid + s]);
        __syncthreads();
    }
    if (tid == 0) Mout[0] = red[0];
}

// =====================================================================
// K3: w[i] = exp(score[i] - M) in place; per-block partial sums (256 blocks)
// =====================================================================
__global__ void k3_exp(float* __restrict__ w,
                       const float* __restrict__ Min,
                       float* __restrict__ psum) {
    __shared__ float red[256];
    const int tid = threadIdx.x;
    const int i   = blockIdx.x * 256 + tid;
    const float M = Min[0];
    const float e = expf(w[i] - M);
    w[i] = e;
    red[tid] = e;
    __syncthreads();
    for (int s = 128; s > 0; s >>= 1) {
        if (tid < s) red[tid] += red[tid + s];
        __syncthreads();
    }
    if (tid == 0) psum[blockIdx.x] = red[0];
}

// =====================================================================
// K4: reduce 256 partial sums -> inv = 1/S  (single block)
// =====================================================================
__global__ void k4_sumreduce(const float* __restrict__ psum,
                             float* __restrict__ inv) {
    __shared__ float red[256];
    const int tid = threadIdx.x;
    red[tid] = psum[tid];
    __syncthreads();
    for (int s = 128; s > 0; s >>= 1) {
        if (tid < s) red[tid] += red[tid + s];
        __syncthreads();
    }
    if (tid == 0) inv[0] = 1.0f / red[0];
}

// =====================================================================
// K5: GEMV  partial[chunk][j] = sum_{c in chunk} w[c] * V[c,j]
//     via V_WMMA_F32_16X16X4_F32.
//     grid: 8192 waves = 128 row-chunks (512 rows) x 64 col-tiles (16 cols).
//     A tile (16x4 f32): row 0 = w[c..c+3], rows 1..15 = 0.
//       lane 0  -> {w[c+0], w[c+1]},  lane 16 -> {w[c+2], w[c+3]}
//     B tile (4x16 f32): VGPR0 = rows K0 (lanes 0-15) / K2 (lanes 16-31),
//                        VGPR1 = rows K1 / K3, at column N = lane&15.
//     D row M=0 = acc[0] on lanes 0-15.
// =====================================================================
__global__ void k5_gemv_wmma(const float* __restrict__ w,
                             const float* __restrict__ V,
                             float* __restrict__ partials) {
    const int wave  = threadIdx.x >> 5;
    const int lane  = threadIdx.x & 31;
    const int gw    = blockIdx.x * 8 + wave;   // 0..8191
    const int chunk = gw >> 6;                 // 0..127 (512 rows each)
    const int tile  = gw & 63;                 // 0..63  (16 cols each)
    const int j0    = tile * 16;
    const int row0  = chunk * CHUNK_ROWS;

    const int col   = j0 + (lane & 15);
    const int rsel  = (lane >> 4) << 1;        // 0 for lanes 0-15, 2 for 16-31
    const bool a0   = (lane == 0);
    const bool a16  = (lane == 16);

    const float* vb = V + (size_t)(row0 + rsel) * OUT_FEAT + col;
    const float* wp = w + row0;

    v8f acc = {};
#pragma unroll 8
    for (int c = 0; c < CHUNK_ROWS; c += 4) {
        // prefetch the values stream 16 rows ahead (clamped in-bounds; uniform cond)
        if (((c & 15) == 0) && (row0 + c + 20 <= CAPACITY))
            __builtin_prefetch(vb + 16 * OUT_FEAT, 0, 3);

        // uniform-address load of 4 weights
        const float4 w4 = *(const float4*)(wp + c);
        v2f a;
        a.x = a0 ? w4.x : (a16 ? w4.z : 0.0f);
        a.y = a0 ? w4.y : (a16 ? w4.w : 0.0f);

        v2f b;
        b.x = vb[0];
        b.y = vb[OUT_FEAT];

        acc = __builtin_amdgcn_wmma_f32_16x16x4_f32(
                  /*neg_a=*/false, a, /*neg_b=*/false, b,
                  /*c_mod=*/(short)0, acc, /*reuse_a=*/false, /*reuse_b=*/false);

        vb += 4 * OUT_FEAT;
    }
    if (lane < 16)
        partials[chunk * OUT_FEAT + j0 + lane] = acc[0];
}

// =====================================================================
// K6: out[j] = inv * sum_{k<128} partials[k][j]   (4 blocks x 256)
// =====================================================================
__global__ void k6_finalize(const float* __restrict__ partials,
                            const float* __restrict__ inv,
                            float* __restrict__ out) {
    const int j = blockIdx.x * 256 + threadIdx.x;
    float s = 0.0f;
#pragma unroll 8
    for (int k = 0; k < CHUNKS; ++k)
        s += partials[k * OUT_FEAT + j];
    out[j] = s * inv[0];
}

extern "C" void kernel_launch(void* const* d_in, const int* in_sizes, int n_in,
                              void* d_out, int out_size, void* d_ws, size_t ws_size,
                              hipStream_t stream) {
    const float* q      = (const float*)d_in[0];   // [1024]
    const float* keys   = (const float*)d_in[1];   // [65536,1024]
    const float* values = (const float*)d_in[2];   // [65536,1024]
    float* out = (float*)d_out;                    // [1024]

    float* ws       = (float*)d_ws;
    float* w        = ws + OFF_W;
    float* bmax     = ws + OFF_BMAX;
    float* M        = ws + OFF_M;
    float* psum     = ws + OFF_PSUM;
    float* inv      = ws + OFF_INV;
    float* partials = ws + OFF_PART;

    k1_scores   <<<CAPACITY / 8, 256, 0, stream>>>(q, keys, w, bmax);
    k2_maxreduce<<<1, 1024, 0, stream>>>(bmax, M);
    k3_exp      <<<CAPACITY / 256, 256, 0, stream>>>(w, M, psum);
    k4_sumreduce<<<1, 256, 0, stream>>>(psum, inv);
    k5_gemv_wmma<<<(CHUNKS * 64) / 8, 256, 0, stream>>>(w, values, partials);
    k6_finalize <<<4, 256, 0, stream>>>(partials, inv, out);
}